// NMSLayer_57166014710240
// MI455X (gfx1250) — compile-verified
//
#include <hip/hip_runtime.h>
#include <hip/hip_bf16.h>
#include <stdint.h>

#define IMG   4096
#define NPIX  (IMG * IMG)
#define KRANK ((NPIX - 1) / 2)   // 8388607

typedef unsigned int v4u __attribute__((ext_vector_type(4)));
typedef int          v8i __attribute__((ext_vector_type(8)));
typedef int          v4i __attribute__((ext_vector_type(4)));

#if defined(__HIP_DEVICE_COMPILE__) && __has_builtin(__builtin_amdgcn_tensor_load_to_lds)
#define HAVE_TDM 1
#else
#define HAVE_TDM 0
#endif

// monotonic float -> uint key (ascending order preserved)
__device__ __forceinline__ unsigned f2key(float f) {
  unsigned u = __float_as_uint(f);
  return (u & 0x80000000u) ? ~u : (u | 0x80000000u);
}

// ---------------------------------------------------------------------------
// workspace zeroing
__global__ __launch_bounds__(256) void zero_ws_kernel(unsigned* p, int n) {
  int i = blockIdx.x * 256 + threadIdx.x;
  if (i < n) p[i] = 0u;
}

// ---------------------------------------------------------------------------
// pass 1: 14-bit (key >> 18) histogram, LDS-privatized per workgroup
__global__ __launch_bounds__(256) void hist_hi_kernel(const float4* __restrict__ x4,
                                                      unsigned* __restrict__ hist1, int n4) {
  __shared__ unsigned lh[16384];
  for (int i = threadIdx.x; i < 16384; i += 256) lh[i] = 0u;
  __syncthreads();
  const int stride = gridDim.x * 256;
  for (int i = blockIdx.x * 256 + threadIdx.x; i < n4; i += stride) {
    float4 v = x4[i];
    atomicAdd(&lh[f2key(v.x) >> 18], 1u);
    atomicAdd(&lh[f2key(v.y) >> 18], 1u);
    atomicAdd(&lh[f2key(v.z) >> 18], 1u);
    atomicAdd(&lh[f2key(v.w) >> 18], 1u);
  }
  __syncthreads();
  for (int i = threadIdx.x; i < 16384; i += 256) {
    unsigned c = lh[i];
    if (c) atomicAdd(&hist1[i], c);
  }
}

// scan hist1 -> bucket b1 (scal[0]) and remaining rank (scal[1])
__global__ __launch_bounds__(256) void scan1_kernel(const unsigned* __restrict__ hist1,
                                                    unsigned* __restrict__ scal) {
  __shared__ unsigned psum[256];
  __shared__ unsigned sel[2];
  const int t = threadIdx.x;
  unsigned s = 0;
  for (int i = 0; i < 64; ++i) s += hist1[t * 64 + i];
  psum[t] = s;
  __syncthreads();
  if (t == 0) {
    unsigned k = (unsigned)KRANK, cum = 0;
    int chunk = 255;
    for (int i = 0; i < 256; ++i) {
      if (cum + psum[i] > k) { chunk = i; break; }
      cum += psum[i];
    }
    sel[0] = (unsigned)chunk; sel[1] = k - cum;
  }
  __syncthreads();
  if (t == (int)sel[0]) {
    unsigned rr = sel[1];
    for (int i = 0; i < 64; ++i) {
      unsigned c = hist1[t * 64 + i];
      if (rr < c) { scal[0] = (unsigned)(t * 64 + i); scal[1] = rr; break; }
      rr -= c;
    }
  }
}

// pass 2: 18-bit (key & 0x3FFFF) histogram of elements inside bucket b1
__global__ __launch_bounds__(256) void hist_lo_kernel(const float4* __restrict__ x4,
                                                      unsigned* __restrict__ hist2,
                                                      const unsigned* __restrict__ scal, int n4) {
  const unsigned b1 = scal[0];
  const int stride = gridDim.x * 256;
  for (int i = blockIdx.x * 256 + threadIdx.x; i < n4; i += stride) {
    float4 v = x4[i];
    unsigned k0 = f2key(v.x), k1 = f2key(v.y), k2 = f2key(v.z), k3 = f2key(v.w);
    if ((k0 >> 18) == b1) atomicAdd(&hist2[k0 & 0x3FFFFu], 1u);
    if ((k1 >> 18) == b1) atomicAdd(&hist2[k1 & 0x3FFFFu], 1u);
    if ((k2 >> 18) == b1) atomicAdd(&hist2[k2 & 0x3FFFFu], 1u);
    if ((k3 >> 18) == b1) atomicAdd(&hist2[k3 & 0x3FFFFu], 1u);
  }
}

// scan hist2 -> exact median bits in scal[2]
__global__ __launch_bounds__(1024) void scan2_kernel(const unsigned* __restrict__ hist2,
                                                     unsigned* __restrict__ scal) {
  __shared__ unsigned psum[1024];
  __shared__ unsigned sel[2];
  const int t = threadIdx.x;
  const unsigned base = (unsigned)t * 256u;
  unsigned s = 0;
  for (int i = 0; i < 256; ++i) s += hist2[base + i];
  psum[t] = s;
  __syncthreads();
  if (t == 0) {
    unsigned k = scal[1], cum = 0;
    int chunk = 1023;
    for (int i = 0; i < 1024; ++i) {
      if (cum + psum[i] > k) { chunk = i; break; }
      cum += psum[i];
    }
    sel[0] = (unsigned)chunk; sel[1] = k - cum;
  }
  __syncthreads();
  if (t == (int)sel[0]) {
    unsigned rr = sel[1];
    for (int i = 0; i < 256; ++i) {
      unsigned c = hist2[base + i];
      if (rr < c) {
        unsigned key = (scal[0] << 18) | (base + (unsigned)i);
        // invert monotonic mapping
        unsigned u = (key & 0x80000000u) ? (key & 0x7FFFFFFFu) : ~key;
        scal[2] = u;
        break;
      }
      rr -= c;
    }
  }
}

// ---------------------------------------------------------------------------
// threshold + 7x7 max-pool + equality-select, 64x64 tile per WG.
// Halo tile staged into LDS with the CDNA5 Tensor Data Mover.
__global__ __launch_bounds__(256) void nms_kernel(const float* __restrict__ x,
                                                  float* __restrict__ out,
                                                  const unsigned* __restrict__ scal) {
  __shared__ float xt[70 * 70];   // raw input halo tile (clamped, row-major HxW)
  __shared__ float hm[70 * 64];   // horizontal 7-max of thresholded values
  __shared__ float smed;

  const int r0 = blockIdx.y * 64, c0 = blockIdx.x * 64;
  const int hr0 = r0 - 3, hc0 = c0 - 3;               // ideal halo origin
  const int srow = hr0 < 0 ? 0 : hr0;
  const int scol = hc0 < 0 ? 0 : hc0;
  const int erow = (r0 + 66 > IMG - 1) ? IMG - 1 : r0 + 66;
  const int ecol = (c0 + 66 > IMG - 1) ? IMG - 1 : c0 + 66;
  const int H = erow - srow + 1;                      // <= 70
  const int W = ecol - scol + 1;                      // <= 70
  const int offY = srow - hr0;                        // 0..3
  const int offX = scol - hc0;                        // 0..3

  if (threadIdx.x == 0) smed = __uint_as_float(scal[2]);

#if HAVE_TDM
  // Issue one TDM 2D tile load (wave 0 only; scalar branch via readfirstlane).
  if (__builtin_amdgcn_readfirstlane((int)threadIdx.x) == 0) {
    unsigned long long gaddr =
        (unsigned long long)(uintptr_t)(x + (size_t)srow * IMG + scol);
    unsigned lds_addr = (unsigned)(uintptr_t)(&xt[0]);
    v4u g0;
    g0.x = 1u;                                          // count=1 (valid), user mode
    g0.y = lds_addr;                                    // LDS byte address
    g0.z = (unsigned)(gaddr & 0xFFFFFFFFu);             // global_addr[31:0]
    g0.w = (unsigned)((gaddr >> 32) & 0x01FFFFFFu) | (2u << 30); // addr[56:32] | type=2
    v8i g1;
    g1[0] = (int)(2u << 16);                 // workgroup_mask=0, data_size=4B, no pad/iter
    g1[1] = (int)(4096u << 16);              // tensor_dim0[15:0] in bits 63:48
    g1[2] = (int)((4096u >> 16) | (4096u << 16)); // dim0[31:16] | tensor_dim1[15:0]
    g1[3] = (int)((4096u >> 16) | ((unsigned)W << 16)); // dim1[31:16] | tile_dim0
    g1[4] = (int)(unsigned)H;                // tile_dim1 | tile_dim2=0
    g1[5] = (int)4096;                       // tensor_dim0_stride[31:0]
    g1[6] = 0;                               // stride0[47:32] | stride1[15:0]
    g1[7] = 0;                               // stride1[47:16]
    v4i gz = {0, 0, 0, 0};
#if __clang_major__ >= 23
    v8i gz8 = {0, 0, 0, 0, 0, 0, 0, 0};
    __builtin_amdgcn_tensor_load_to_lds(g0, g1, gz, gz, gz8, 0);
#else
    __builtin_amdgcn_tensor_load_to_lds(g0, g1, gz, gz, 0);
#endif
  }
  __builtin_amdgcn_s_wait_tensorcnt((unsigned short)0);
#else
  // fallback cooperative staging (also used for the host-side parse)
  for (int i = threadIdx.x; i < H * W; i += 256) {
    int rr = i / W, cc = i - rr * W;
    xt[i] = x[(size_t)(srow + rr) * IMG + (scol + cc)];
  }
#endif
  __syncthreads();

  const float med = smed;
  const float NEG_INF = -__builtin_inff();
  const int c  = threadIdx.x & 63;   // output column within tile
  const int rb = threadIdx.x >> 6;   // 0..3 row group

  // horizontal pass: hmax over thresholded values (-inf outside image)
  for (int hr = rb; hr < 70; hr += 4) {
    float m = NEG_INF;
    if (hr >= offY && hr < offY + H) {
      const int lr = hr - offY;
#pragma unroll
      for (int d = 0; d < 7; ++d) {
        const int hc = c + d;                 // halo column 0..69
        float tv = NEG_INF;
        if (hc >= offX && hc < offX + W) {
          float v = xt[lr * W + (hc - offX)];
          tv = (v < med) ? 0.0f : v;
        }
        m = fmaxf(m, tv);
      }
    }
    hm[hr * 64 + c] = m;
  }
  __syncthreads();

  // vertical pass + output
  for (int r = rb; r < 64; r += 4) {
    float pooled = NEG_INF;
#pragma unroll
    for (int i = 0; i < 7; ++i) pooled = fmaxf(pooled, hm[(r + i) * 64 + c]);
    const float xv = xt[(r + 3 - offY) * W + (c + 3 - offX)];   // center (always in-image)
    const float tc = (xv < med) ? 0.0f : xv;
    out[(size_t)(r0 + r) * IMG + (c0 + c)] = (tc == pooled) ? xv : 0.0f;
  }
}

// ---------------------------------------------------------------------------
extern "C" void kernel_launch(void* const* d_in, const int* in_sizes, int n_in,
                              void* d_out, int out_size, void* d_ws, size_t ws_size,
                              hipStream_t stream) {
  (void)in_sizes; (void)n_in; (void)out_size; (void)ws_size;
  const float* x = (const float*)d_in[0];
  float* out = (float*)d_out;

  unsigned* hist1 = (unsigned*)d_ws;           // 16384 u32
  unsigned* hist2 = hist1 + 16384;             // 262144 u32
  unsigned* scal  = hist2 + 262144;            // [0]=b1 [1]=rank [2]=median bits

  const int n4 = NPIX / 4;
  const int zn = 16384 + 262144 + 8;

  zero_ws_kernel<<<(zn + 255) / 256, 256, 0, stream>>>((unsigned*)d_ws, zn);
  hist_hi_kernel<<<512, 256, 0, stream>>>((const float4*)x, hist1, n4);
  scan1_kernel<<<1, 256, 0, stream>>>(hist1, scal);
  hist_lo_kernel<<<2048, 256, 0, stream>>>((const float4*)x, hist2, scal, n4);
  scan2_kernel<<<1, 1024, 0, stream>>>(hist2, scal);

  dim3 grid(IMG / 64, IMG / 64, 1);
  nms_kernel<<<grid, 256, 0, stream>>>(x, out, scal);
}